// GraphSAGE_68453188763778
// MI455X (gfx1250) — compile-verified
//
#include <hip/hip_runtime.h>

// GraphSAGE (3x SAGEConv mean-aggr + linear head) for gfx1250 / MI455X.
// fp32 end-to-end. GEMMs on the matrix pipe via V_WMMA_F32_16X16X4_F32 with
// LDS-staged weight fragments; aggregation via CSR gather (no float atomics).

typedef __attribute__((ext_vector_type(2))) float v2f;
typedef __attribute__((ext_vector_type(8))) float v8f;

#define FEAT 128
#define TILE_M 16
#define KS_STAGE 16                 // k-steps (of 4) staged in LDS per stage
#define FRAGS_PER_STAGE (KS_STAGE * 8 * 32)   // 4096 v2f = 32KB per matrix

// ---------------- utility kernels ----------------

__global__ void sage_zero_f32x4(float4* __restrict__ p, long n4) {
    long i = (long)blockIdx.x * blockDim.x + threadIdx.x;
    long stride = (long)gridDim.x * blockDim.x;
    float4 z = make_float4(0.f, 0.f, 0.f, 0.f);
    for (; i < n4; i += stride) p[i] = z;
}

__global__ void sage_degree(const int* __restrict__ ei, int* __restrict__ deg, int E) {
    int i = blockIdx.x * blockDim.x + threadIdx.x;
    int stride = gridDim.x * blockDim.x;
    for (; i < E; i += stride) atomicAdd(&deg[ei[E + i]], 1);
}

__global__ void sage_invdeg(const int* __restrict__ deg, float* __restrict__ inv, int N) {
    int i = blockIdx.x * blockDim.x + threadIdx.x;
    if (i < N) {
        int d = deg[i];
        inv[i] = 1.0f / (float)(d > 1 ? d : 1);
    }
}

// Single-block exclusive scan (Hillis-Steele, 1024 threads, N/1024 chunks with carry).
__global__ void sage_scan(const int* __restrict__ deg, int* __restrict__ rowptr,
                          int* __restrict__ cursor, int N) {
    __shared__ int tmp[1024];
    __shared__ int carry;
    int tid = threadIdx.x;
    if (tid == 0) carry = 0;
    __syncthreads();
    for (int base = 0; base < N; base += 1024) {
        int i = base + tid;
        int v = (i < N) ? deg[i] : 0;
        int c = carry;
        int s = v;
        tmp[tid] = s;
        __syncthreads();
        for (int off = 1; off < 1024; off <<= 1) {
            int a = (tid >= off) ? tmp[tid - off] : 0;
            __syncthreads();
            s += a;
            tmp[tid] = s;
            __syncthreads();
        }
        if (i < N) {
            int excl = c + s - v;
            rowptr[i] = excl;
            cursor[i] = excl;
        }
        int tot = tmp[1023];
        __syncthreads();
        if (tid == 0) carry = c + tot;
        __syncthreads();
    }
    if (tid == 0) rowptr[N] = carry;
}

__global__ void sage_fill(const int* __restrict__ ei, int* __restrict__ cursor,
                          int* __restrict__ col, int E) {
    int i = blockIdx.x * blockDim.x + threadIdx.x;
    int stride = gridDim.x * blockDim.x;
    for (; i < E; i += stride) {
        int dst = ei[E + i];
        int pos = atomicAdd(&cursor[dst], 1);
        col[pos] = ei[i];
    }
}

// CSR gather-sum: one wave per node, lane owns 4 contiguous features.
// Features are L2-resident (51MB << 192MB); each neighbor row read is a
// coalesced 512B wave transaction; single plain 512B store per node.
__global__ void sage_gather(const float* __restrict__ h, const int* __restrict__ rowptr,
                            const int* __restrict__ col, float* __restrict__ agg, int N) {
    int lane  = threadIdx.x & 31;
    int gwave = (blockIdx.x * blockDim.x + threadIdx.x) >> 5;
    int nwave = (gridDim.x * blockDim.x) >> 5;
    for (int i = gwave; i < N; i += nwave) {
        int beg = rowptr[i], end = rowptr[i + 1];
        float4 s = make_float4(0.f, 0.f, 0.f, 0.f);
        for (int j = beg; j < end; ++j) {
            int src = col[j];
            if (j + 1 < end)  // prefetch next neighbor row (global_prefetch_b8)
                __builtin_prefetch(h + (size_t)col[j + 1] * FEAT + lane * 4, 0, 0);
            const float4 v = *(const float4*)(h + (size_t)src * FEAT + lane * 4);
            s.x += v.x; s.y += v.y; s.z += v.z; s.w += v.w;
        }
        *(float4*)(agg + (size_t)i * FEAT + lane * 4) = s;
    }
}

// ------- fused SAGE linear: relu(inv*agg @ wl^T + bl + h @ wr^T) via f32 WMMA -------
//
// Block = 256 threads (8 waves); each wave owns 2 node tiles (32 nodes x 128 out).
// Weights are staged per-K-half into LDS in *fragment order* [ks][t][lane] so the
// inner-loop B fetch is a single conflict-free ds_load_b64 per fragment.
// f32 WMMA layouts (ISA 7.12.2): A lane r holds row M=r, VGPR pair = K(2*half),K(2*half+1);
// B lane r holds col N=r with same K split; C/D: VGPR v -> row v+8*half, col r.
__global__ __launch_bounds__(256) void sage_linear_wmma(
        const float* __restrict__ hin, const float* __restrict__ agg,
        const float* __restrict__ inv, const float* __restrict__ wl,
        const float* __restrict__ bl,  const float* __restrict__ wr,
        float* __restrict__ hout, int n_pairs) {
    __shared__ v2f sWL[FRAGS_PER_STAGE];   // 32KB
    __shared__ v2f sWR[FRAGS_PER_STAGE];   // 32KB  (total 64KB)

    int tid  = threadIdx.x;
    int lane = tid & 31;
    int wid  = tid >> 5;
    int half = lane >> 4;
    int r    = lane & 15;

    int pair   = blockIdx.x * 8 + wid;
    bool active = pair < n_pairs;
    int cp     = active ? pair : 0;        // clamp so loads stay in-bounds
    int node0  = cp * 2 * TILE_M;          // tile 0 of this wave
    int node1  = node0 + TILE_M;           // tile 1

    float sc0 = inv[node0 + r];
    float sc1 = inv[node1 + r];
    const float* aggR0 = agg + (size_t)(node0 + r) * FEAT;
    const float* hinR0 = hin + (size_t)(node0 + r) * FEAT;
    const float* aggR1 = agg + (size_t)(node1 + r) * FEAT;
    const float* hinR1 = hin + (size_t)(node1 + r) * FEAT;

    v8f acc0[8], acc1[8];
    const v8f vzero = {};
#pragma unroll
    for (int t = 0; t < 8; ++t) { acc0[t] = vzero; acc1[t] = vzero; }

    for (int stage = 0; stage < 2; ++stage) {
        int kbase = stage * KS_STAGE * 4;
        // stage weight fragments: idx -> (ks = idx>>8, t = (idx>>5)&7, lane = idx&31)
        for (int idx = tid; idx < FRAGS_PER_STAGE; idx += 256) {
            int l  = idx & 31;
            int tt = (idx >> 5) & 7;
            int ks = idx >> 8;
            int kk = kbase + ks * 4 + 2 * (l >> 4);
            size_t row = (size_t)(tt * 16 + (l & 15)) * FEAT + kk;
            sWL[idx] = *(const v2f*)(wl + row);
            sWR[idx] = *(const v2f*)(wr + row);
        }
        __syncthreads();

        for (int ks = 0; ks < KS_STAGE; ++ks) {
            int k = kbase + ks * 4 + 2 * half;
            v2f aA0 = *(const v2f*)(aggR0 + k);
            v2f aX0 = *(const v2f*)(hinR0 + k);
            v2f aA1 = *(const v2f*)(aggR1 + k);
            v2f aX1 = *(const v2f*)(hinR1 + k);
            aA0.x *= sc0; aA0.y *= sc0;
            aA1.x *= sc1; aA1.y *= sc1;

            v2f bL[8], bR[8];
#pragma unroll
            for (int t = 0; t < 8; ++t) bL[t] = sWL[(ks * 8 + t) * 32 + lane];
#pragma unroll
            for (int t = 0; t < 8; ++t) bR[t] = sWR[(ks * 8 + t) * 32 + lane];

#pragma unroll
            for (int t = 0; t < 8; ++t) {
                acc0[t] = __builtin_amdgcn_wmma_f32_16x16x4_f32(
                    false, aA0, false, bL[t], (short)0, acc0[t], false, false);
                acc0[t] = __builtin_amdgcn_wmma_f32_16x16x4_f32(
                    false, aX0, false, bR[t], (short)0, acc0[t], false, false);
                acc1[t] = __builtin_amdgcn_wmma_f32_16x16x4_f32(
                    false, aA1, false, bL[t], (short)0, acc1[t], false, false);
                acc1[t] = __builtin_amdgcn_wmma_f32_16x16x4_f32(
                    false, aX1, false, bR[t], (short)0, acc1[t], false, false);
            }
        }
        __syncthreads();   // protect LDS before next stage overwrites it
    }

    if (active) {
#pragma unroll
        for (int t = 0; t < 8; ++t) {
            int col = t * 16 + r;
            float bias = bl[col];
#pragma unroll
            for (int v = 0; v < 8; ++v) {
                float v0 = fmaxf(acc0[t][v] + bias, 0.0f);
                float v1 = fmaxf(acc1[t][v] + bias, 0.0f);
                hout[(size_t)(node0 + v + 8 * half) * FEAT + col] = v0;
                hout[(size_t)(node1 + v + 8 * half) * FEAT + col] = v1;
            }
        }
    }
}

// ---------------- head: out[i] = h[i] . w_out + b_out ----------------

__global__ void sage_head(const float* __restrict__ h, const float* __restrict__ w,
                          const float* __restrict__ b, float* __restrict__ out, int N) {
    int lane  = threadIdx.x & 31;
    int gwave = (blockIdx.x * blockDim.x + threadIdx.x) >> 5;
    int nwave = (gridDim.x * blockDim.x) >> 5;
    for (int i = gwave; i < N; i += nwave) {
        const float4 x  = *(const float4*)(h + (size_t)i * FEAT + lane * 4);
        const float4 wv = *(const float4*)(w + lane * 4);
        float s = x.x * wv.x + x.y * wv.y + x.z * wv.z + x.w * wv.w;
#pragma unroll
        for (int off = 16; off > 0; off >>= 1) s += __shfl_down(s, off, 32);
        if (lane == 0) out[i] = s + b[0];
    }
}

// ---------------- launch ----------------

extern "C" void kernel_launch(void* const* d_in, const int* in_sizes, int n_in,
                              void* d_out, int out_size, void* d_ws, size_t ws_size,
                              hipStream_t stream) {
    const float* x     = (const float*)d_in[0];
    const int*   ei    = (const int*)d_in[1];
    const float* wl[3] = {(const float*)d_in[2], (const float*)d_in[5], (const float*)d_in[8]};
    const float* blv[3]= {(const float*)d_in[3], (const float*)d_in[6], (const float*)d_in[9]};
    const float* wr[3] = {(const float*)d_in[4], (const float*)d_in[7], (const float*)d_in[10]};
    const float* w_out = (const float*)d_in[11];
    const float* b_out = (const float*)d_in[12];

    const int N = in_sizes[0] / FEAT;   // 100000 (multiple of 32)
    const int E = in_sizes[1] / 2;      // 1.6M

    // workspace layout (256B-aligned chunks)
    char* ws = (char*)d_ws;
    size_t off = 0;
    auto take = [&](size_t bytes) {
        char* p = ws + off;
        off += (bytes + 255) & ~(size_t)255;
        return p;
    };
    size_t featBytes = (size_t)N * FEAT * sizeof(float);   // 51.2 MB
    float* hA     = (float*)take(featBytes);
    float* hB     = (float*)take(featBytes);
    float* agg    = (float*)take(featBytes);
    int*   deg    = (int*)  take((size_t)N * 4);
    float* inv    = (float*)take((size_t)N * 4);
    int*   rowptr = (int*)  take((size_t)(N + 1) * 4);
    int*   cursor = (int*)  take((size_t)N * 4);
    int*   col    = (int*)  take((size_t)E * 4);

    dim3 blk(256);

    // ---- build CSR once (edge list is layer-invariant) ----
    sage_zero_f32x4<<<256, blk, 0, stream>>>((float4*)deg, (long)(N / 4));
    sage_degree<<<2048, blk, 0, stream>>>(ei, deg, E);
    sage_invdeg<<<(N + 255) / 256, blk, 0, stream>>>(deg, inv, N);
    sage_scan<<<1, 1024, 0, stream>>>(deg, rowptr, cursor, N);
    sage_fill<<<2048, blk, 0, stream>>>(ei, cursor, col, E);

    // ---- 3 SAGE layers ----
    const float* hin = x;
    float* houts[3] = {hA, hB, hA};
    int n_pairs = N / (2 * TILE_M);                 // 3125
    int linBlocks = (n_pairs + 7) / 8;              // 8 waves/block, 1 pair/wave

    for (int l = 0; l < 3; ++l) {
        sage_gather<<<4096, blk, 0, stream>>>(hin, rowptr, col, agg, N);
        sage_linear_wmma<<<linBlocks, blk, 0, stream>>>(hin, agg, inv,
                                                        wl[l], blv[l], wr[l],
                                                        houts[l], n_pairs);
        hin = houts[l];
    }

    sage_head<<<2048, blk, 0, stream>>>(hin, w_out, b_out, (float*)d_out, N);
}